// MiniRocketClassifier_9835475108408
// MI455X (gfx1250) — compile-verified
//
#include <hip/hip_runtime.h>

// ---------------------------------------------------------------------------
// MiniRocket feature extraction on gfx1250 (MI455X), WMMA-f32 formulation.
//
//   resp[b,d,k,l] = sum_{j,c} (kern[k,j]*mask[d,k,c]) * x[b, l+(j-4)*dil, c]
//                 = W_d[96x72] @ patch[72 x L]          (K padded 84 -> 96)
//   feat[b,d,k,f] = sum_l w[k,l] * (resp > bias[d,k,f])
//   out           = (feat - mean) / std
//
// One workgroup per (batch, dilation); 8 waves; V_WMMA_F32_16X16X4_F32 with
// contraction 72 = 18 K-steps. Body templated on dilation index so every
// LDS B-gather is a ds_load_b64 with an immediate offset off one base VGPR.
// ---------------------------------------------------------------------------

#define BDIM 256

constexpr int Bsz  = 64;
constexpr int Lsz  = 2048;
constexpr int Csz  = 8;
constexpr int Dsz  = 4;
constexpr int Ksz  = 84;
constexpr int KPAD = 96;          // 6 M-tiles of 16
constexpr int Fsz  = 4;
constexpr int Jsz  = 9;
constexpr int RED  = Jsz * Csz;   // 72 contraction
constexpr int NS   = RED / 4;     // 18 WMMA K-steps
constexpr int PADH = 32;          // halo: max |(j-4)*dil| = 4*8
constexpr int LPAD = Lsz + 2 * PADH;
constexpr int NFEAT = Dsz * Ksz * Fsz;   // 1344
constexpr int SMEM_FLOATS = LPAD * Csz + KPAD * RED + KPAD * Fsz;

typedef __attribute__((ext_vector_type(2))) float v2f;
typedef __attribute__((ext_vector_type(8))) float v8f;

template <int DIDX>
__device__ __forceinline__ void mr_body(
    const float* __restrict__ x,        // [B, L, C]
    const float* __restrict__ kernels,  // [K, 9]
    const float* __restrict__ masks,    // [D, K, C]
    const float* __restrict__ biases,   // [D, K, F]
    const float* __restrict__ fmean,    // [D*K*F]
    const float* __restrict__ fstd,     // [D*K*F]
    float* __restrict__ out,            // [B, D*K*F]
    float* sX, float* sW, float* sF, int b)
{
    constexpr int   DIL  = 1 << DIDX;
    constexpr int   PAD  = 4 * DIL;
    constexpr float invL = 1.0f / (float)Lsz;
    constexpr float invT = 1.0f / (float)(Lsz - 2 * PAD);

    const int tid = threadIdx.x;

    // ---- init: zero halos + feature accumulators -------------------------
    for (int i = tid; i < PADH * Csz; i += BDIM) {
        sX[i] = 0.0f;
        sX[(PADH + Lsz) * Csz + i] = 0.0f;
    }
    for (int i = tid; i < KPAD * Fsz; i += BDIM) sF[i] = 0.0f;

    // ---- stage x[b] into LDS (float4 copies) -----------------------------
    {
        const float4* xg = (const float4*)(x + (size_t)b * Lsz * Csz);
        float4* xs = (float4*)(sX + PADH * Csz);
        for (int i = tid; i < (Lsz * Csz) / 4; i += BDIM) xs[i] = xg[i];
    }

    // ---- build fused weights W[k][j*8+c] = kern[k][j] * mask[d][k][c] ----
    for (int i = tid; i < KPAD * RED; i += BDIM) {
        const int k = i / RED;
        const int r = i - k * RED;
        const int j = r >> 3;
        const int c = r & 7;
        float w = 0.0f;
        if (k < Ksz) w = kernels[k * Jsz + j] * masks[(DIDX * Ksz + k) * Csz + c];
        sW[i] = w;
    }
    __syncthreads();

    const int wave = tid >> 5;
    const int lane = tid & 31;
    const int lo   = lane & 15;   // N position / A row
    const int hi   = lane >> 4;   // half-wave select

    for (int m = 0; m < KPAD / 16; ++m) {
        // A fragments for all 18 K-steps (lane = row m*16+lo; K = 4s + v + 2*hi)
        v2f afrag[NS];
        const float* wrow = sW + (m * 16 + lo) * RED + 2 * hi;
        #pragma unroll
        for (int s = 0; s < NS; ++s)
            afrag[s] = *(const v2f*)(wrow + 4 * s);

        // per-lane biases for this tile's 8 C/D rows (k = m*16 + r + 8*hi)
        float4 brow[8];
        #pragma unroll
        for (int r = 0; r < 8; ++r) {
            const int k = m * 16 + r + 8 * hi;
            brow[r] = (k < Ksz)
                ? *(const float4*)(biases + (size_t)(DIDX * Ksz + k) * Fsz)
                : make_float4(3.0e38f, 3.0e38f, 3.0e38f, 3.0e38f);
        }

        float acc[8][4] = {};

        // each wave owns 16 of the 128 L-tiles
        for (int lt = wave; lt < Lsz / 16; lt += BDIM / 32) {
            const int l = lt * 16 + lo;          // this lane's N position
            // B-gather base: hi folds into the base; -256 bias keeps all
            // per-step offsets non-negative immediates.
            const float* p = sX + (PADH + l) * Csz + 2 * hi - 256;

            // Load all 18 B fragments first (batched ds_load_b64, few waits)
            v2f bfrag[NS];
            #pragma unroll
            for (int s = 0; s < NS; ++s) {
                // K-row kk = 4s + v + 2*hi -> tap j = s>>1 (hi-independent),
                // channel c = 4*(s&1) + 2*hi (+v). Compile-time offset:
                const int off = ((s >> 1) - 4) * DIL * Csz + 4 * (s & 1) + 256;
                bfrag[s] = *(const v2f*)(p + off);
            }

            // 18 back-to-back WMMAs (accumulator chain is HW-pipelined)
            v8f cacc = {};
            #pragma unroll
            for (int s = 0; s < NS; ++s)
                cacc = __builtin_amdgcn_wmma_f32_16x16x4_f32(
                    false, afrag[s], false, bfrag[s], (short)0, cacc, false, false);

            // PPV accumulate: weight = parity-odd ? trimmed interior : 1/L
            const float wtrim = (l >= PAD && l < Lsz - PAD) ? invT : 0.0f;
            #pragma unroll
            for (int r = 0; r < 8; ++r) {
                const float wsel = ((DIDX + r) & 1) ? wtrim : invL;
                const float v = cacc[r];
                acc[r][0] += (v > brow[r].x) ? wsel : 0.0f;
                acc[r][1] += (v > brow[r].y) ? wsel : 0.0f;
                acc[r][2] += (v > brow[r].z) ? wsel : 0.0f;
                acc[r][3] += (v > brow[r].w) ? wsel : 0.0f;
            }
        }

        // reduce across the 16 N-lanes of each half, merge into sF
        #pragma unroll
        for (int r = 0; r < 8; ++r) {
            #pragma unroll
            for (int f = 0; f < 4; ++f) {
                float v = acc[r][f];
                v += __shfl_xor(v, 1, 16);
                v += __shfl_xor(v, 2, 16);
                v += __shfl_xor(v, 4, 16);
                v += __shfl_xor(v, 8, 16);
                if (lo == 0)
                    atomicAdd(&sF[(m * 16 + r + 8 * hi) * Fsz + f], v);
            }
        }
    }
    __syncthreads();

    // ---- normalize + write out[b, d*K*F + k*F + f] -----------------------
    for (int i = tid; i < Ksz * Fsz; i += BDIM) {
        const int gf = DIDX * Ksz * Fsz + i;
        out[(size_t)b * NFEAT + gf] = (sF[i] - fmean[gf]) / fstd[gf];
    }
}

__global__ __launch_bounds__(BDIM) void minirocket_wmma_kernel(
    const float* __restrict__ x, const float* __restrict__ kernels,
    const float* __restrict__ masks, const float* __restrict__ biases,
    const float* __restrict__ fmean, const float* __restrict__ fstd,
    float* __restrict__ out)
{
    extern __shared__ float smem[];
    float* sX = smem;                    // [LPAD][C]
    float* sW = sX + LPAD * Csz;         // [KPAD][RED]
    float* sF = sW + KPAD * RED;         // [KPAD][F]

    const int b = blockIdx.x >> 2;
    const int d = blockIdx.x & 3;

    switch (d) {  // uniform per block; gives compile-time dilation
    case 0:  mr_body<0>(x, kernels, masks, biases, fmean, fstd, out, sX, sW, sF, b); break;
    case 1:  mr_body<1>(x, kernels, masks, biases, fmean, fstd, out, sX, sW, sF, b); break;
    case 2:  mr_body<2>(x, kernels, masks, biases, fmean, fstd, out, sX, sW, sF, b); break;
    default: mr_body<3>(x, kernels, masks, biases, fmean, fstd, out, sX, sW, sF, b); break;
    }
}

extern "C" void kernel_launch(void* const* d_in, const int* in_sizes, int n_in,
                              void* d_out, int out_size, void* d_ws, size_t ws_size,
                              hipStream_t stream) {
    (void)in_sizes; (void)n_in; (void)d_ws; (void)ws_size; (void)out_size;
    const float* x       = (const float*)d_in[0];
    const float* kernels = (const float*)d_in[1];
    const float* masks   = (const float*)d_in[2];
    const float* biases  = (const float*)d_in[3];
    const float* fmean   = (const float*)d_in[4];
    const float* fstd    = (const float*)d_in[5];
    float* out = (float*)d_out;

    const size_t smem_bytes = (size_t)SMEM_FLOATS * sizeof(float); // ~94.5 KB
    minirocket_wmma_kernel<<<Bsz * Dsz, BDIM, smem_bytes, stream>>>(
        x, kernels, masks, biases, fmean, fstd, out);
}